// GCN_6657199309576
// MI455X (gfx1250) — compile-verified
//
#include <hip/hip_runtime.h>
#include <hip/hip_bf16.h>

typedef __attribute__((ext_vector_type(2))) float v2f;
typedef __attribute__((ext_vector_type(8))) float v8f;

#define N_NODES 100000
#define HIDDEN  128
#define N_EDGES 600000

// ---------------------------------------------------------------- utilities
__global__ void __launch_bounds__(256) fill_zero_kernel(float* __restrict__ p, int n) {
  int i = blockIdx.x * blockDim.x + threadIdx.x;
  if (i < n) p[i] = 0.0f;
}

// Initialize aggregation buffer with the per-feature bias (replaces the
// separate zero-fill + bias-add passes: saves 1 full R+W sweep per layer).
__global__ void __launch_bounds__(256) fill_bias_kernel(float* __restrict__ p,
                                                        const float* __restrict__ b) {
  int i = blockIdx.x * blockDim.x + threadIdx.x;
  if (i < N_NODES * HIDDEN) p[i] = b[i & (HIDDEN - 1)];
}

__global__ void __launch_bounds__(256) deg_kernel(const long long* __restrict__ col,
                                                  float* __restrict__ deg) {
  int e = blockIdx.x * blockDim.x + threadIdx.x;
  if (e < N_EDGES) atomicAdd(&deg[(int)col[e]], 1.0f);
}

__global__ void __launch_bounds__(256) dinv_kernel(float* __restrict__ deg) {
  int i = blockIdx.x * blockDim.x + threadIdx.x;
  if (i < N_NODES) {
    float d = deg[i];
    deg[i] = (d > 0.0f) ? rsqrtf(fmaxf(d, 1.0f)) : 0.0f;
  }
}

__global__ void __launch_bounds__(256) norm_kernel(const long long* __restrict__ row,
                                                   const long long* __restrict__ col,
                                                   const float* __restrict__ dinv,
                                                   float* __restrict__ norm) {
  int e = blockIdx.x * blockDim.x + threadIdx.x;
  if (e < N_EDGES) norm[e] = dinv[(int)row[e]] * dinv[(int)col[e]];
}

// ---------------------------------------------------------------- WMMA GEMM
// H[M x 128] = act(X)[M x 128] @ W[128 x 128]  (fp32 V_WMMA_F32_16X16X4_F32)
// RELU_IN applies the previous layer's ReLU lazily on the A-operand load.
// One wave computes one 16x16 tile; block = 8 waves covering all N=128.
// M = 100000 = 6250 * 16, so no tail -> EXEC is all-ones (WMMA requirement).
template <bool RELU_IN>
__global__ void __launch_bounds__(256) gemm_wmma_kernel(const float* __restrict__ X,
                                                        const float* __restrict__ W,
                                                        float* __restrict__ H) {
  const int wave = threadIdx.x >> 5;   // 0..7 -> N tile
  const int lane = threadIdx.x & 31;
  const int m0   = blockIdx.x << 4;    // 16 rows per block
  const int n0   = wave << 4;
  const int l15  = lane & 15;
  const int hi   = lane >> 4;          // 0 or 1 (selects K pair within step)

  // A-frag: lane l -> row m0+l15, VGPR0/1 = K = 2*hi + (0,1) within each 4-step
  const float* xrow = X + (size_t)(m0 + l15) * HIDDEN + (hi << 1);
  // B-frag: lane l -> col n0+l15, VGPR0/1 = K = 2*hi + (0,1)
  const float* wcol = W + (size_t)(hi << 1) * HIDDEN + (n0 + l15);

  v8f c = {};
#pragma unroll 4
  for (int k = 0; k < HIDDEN; k += 4) {
    v2f a = *(const v2f*)(xrow + k);
    if (RELU_IN) {
      a.x = fmaxf(a.x, 0.0f);
      a.y = fmaxf(a.y, 0.0f);
    }
    v2f b;
    b.x = wcol[k * HIDDEN];
    b.y = wcol[k * HIDDEN + HIDDEN];
    // 8 args: (neg_a, A, neg_b, B, c_mod, C, reuse_a, reuse_b)
    c = __builtin_amdgcn_wmma_f32_16x16x4_f32(false, a, false, b,
                                              (short)0, c, false, false);
  }

  // C/D layout: VGPR i -> M = m0 + i + 8*hi, lane l15 -> N = n0 + l15
  float* hrow = H + (size_t)(m0 + (hi << 3)) * HIDDEN + (n0 + l15);
#pragma unroll
  for (int i = 0; i < 8; ++i) hrow[(size_t)i * HIDDEN] = c[i];
}

// ---------------------------------------------------------------- edge scatter
// One wave per edge: gather 128-float source row (float4 per lane, coalesced
// 512B), scale by norm, scatter-add into target row with non-returning f32
// atomics (resolved in the 192MB L2 where H and agg both fit). Edge metadata
// is wave-uniform -> readfirstlane so it comes in via scalar loads.
__global__ void __launch_bounds__(256) scatter_kernel(const long long* __restrict__ row,
                                                      const long long* __restrict__ col,
                                                      const float* __restrict__ norm,
                                                      const float* __restrict__ H,
                                                      float* __restrict__ agg) {
  int e = (blockIdx.x << 3) + (threadIdx.x >> 5);
  e = __builtin_amdgcn_readfirstlane(e);   // wave-uniform -> SGPR/scalar loads
  if (e >= N_EDGES) return;
  const int lane = threadIdx.x & 31;
  const int src  = (int)row[e];
  const int dst  = (int)col[e];
  const float w  = norm[e];

  const float4 v = *(const float4*)(H + (size_t)src * HIDDEN + (lane << 2));
  float* out = agg + (size_t)dst * HIDDEN + (lane << 2);
  atomicAdd(out + 0, v.x * w);
  atomicAdd(out + 1, v.y * w);
  atomicAdd(out + 2, v.z * w);
  atomicAdd(out + 3, v.w * w);
}

// ---------------------------------------------------------------- launch
extern "C" void kernel_launch(void* const* d_in, const int* in_sizes, int n_in,
                              void* d_out, int out_size, void* d_ws, size_t ws_size,
                              hipStream_t stream) {
  const float* x  = (const float*)d_in[0];
  const float* W1 = (const float*)d_in[1];
  const float* b1 = (const float*)d_in[2];
  const float* W2 = (const float*)d_in[3];
  const float* b2 = (const float*)d_in[4];
  const float* W3 = (const float*)d_in[5];
  const float* b3 = (const float*)d_in[6];
  const long long* edge = (const long long*)d_in[7];   // int64 [2, E]
  const long long* erow = edge;
  const long long* ecol = edge + N_EDGES;

  float* out = (float*)d_out;

  // workspace layout (floats): deg/dinv [100000] | norm [600000] | H [12.8M]
  float* deg  = (float*)d_ws;
  float* norm = deg + N_NODES;
  float* H    = norm + N_EDGES;          // byte offset 2.8MB, 16B-aligned

  const int T = 256;
  const int nodeElems = N_NODES * HIDDEN;

  // degree -> dinv_sqrt -> per-edge norm
  fill_zero_kernel<<<(N_NODES + T - 1) / T, T, 0, stream>>>(deg, N_NODES);
  deg_kernel<<<(N_EDGES + T - 1) / T, T, 0, stream>>>(ecol, deg);
  dinv_kernel<<<(N_NODES + T - 1) / T, T, 0, stream>>>(deg);
  norm_kernel<<<(N_EDGES + T - 1) / T, T, 0, stream>>>(erow, ecol, deg, norm);

  const dim3 gemmGrid(N_NODES / 16);      // 6250
  const dim3 scatGrid(N_EDGES / 8);       // 75000, one wave per edge
  const int  nodeBlocks = (nodeElems + T - 1) / T;

  // ---- layer 1: out = scatter(X @ W1) + b1        (ReLU deferred to GEMM 2)
  gemm_wmma_kernel<false><<<gemmGrid, T, 0, stream>>>(x, W1, H);
  fill_bias_kernel<<<nodeBlocks, T, 0, stream>>>(out, b1);
  scatter_kernel<<<scatGrid, T, 0, stream>>>(erow, ecol, norm, H, out);

  // ---- layer 2: out = scatter(relu(out) @ W2) + b2 (ReLU deferred to GEMM 3)
  gemm_wmma_kernel<true><<<gemmGrid, T, 0, stream>>>(out, W2, H);
  fill_bias_kernel<<<nodeBlocks, T, 0, stream>>>(out, b2);
  scatter_kernel<<<scatGrid, T, 0, stream>>>(erow, ecol, norm, H, out);

  // ---- layer 3: out = scatter(relu(out) @ W3) + b3 (no final ReLU)
  gemm_wmma_kernel<true><<<gemmGrid, T, 0, stream>>>(out, W3, H);
  fill_bias_kernel<<<nodeBlocks, T, 0, stream>>>(out, b3);
  scatter_kernel<<<scatGrid, T, 0, stream>>>(erow, ecol, norm, H, out);
}